// Module_43619687858314
// MI455X (gfx1250) — compile-verified
//
#include <hip/hip_runtime.h>
#include <hip/hip_bf16.h>

typedef __attribute__((ext_vector_type(2))) float v2f;
typedef __attribute__((ext_vector_type(8))) float v8f;

#define N_USERS 100000
#define N_ITEMS 100000
#define DIM 64
#define B_BATCH 4096
#define L_HIST 200
#define TAU 4.0f
#define BETA 0.25f
#define EPSF 1e-8f

__device__ __forceinline__ unsigned mix32(unsigned x) {
    x ^= x >> 16; x *= 0x7feb352du;
    x ^= x >> 15; x *= 0x846ca68bu;
    x ^= x >> 16; return x;
}

// ---------------------------------------------------------------------------
// K0: zero the 4 KL accumulators (ws is poisoned with 0xAA by the harness)
// ---------------------------------------------------------------------------
__global__ void init_acc_kernel(float* __restrict__ acc) {
    if (threadIdx.x < 4) acc[threadIdx.x] = 0.0f;
}

// ---------------------------------------------------------------------------
// Shared GEMM helpers: fp32 WMMA 16x16x4, K=64.
// B staged in LDS in *fragment order*: Bs[p*128 + n*2 + j] = W[(2p+j)*64 + n]
// so each lane's B fragment (K=k0,k0+1 at column n) is one contiguous b64 read.
//   A frag (16x4 f32): lane = 16*half + m ; vgpr0/1 = K = k0 + {0,1}, k0 = kt*4+half*2
//   C/D   (16x16 f32): vgpr r, lane l -> M = r + 8*(l>>4), N = l&15
// Each wave computes a 32-row tile (two 16x64 blocks sharing B fragments).
// ---------------------------------------------------------------------------
__device__ __forceinline__ void stage_B_frag(const float* __restrict__ W,
                                             float* __restrict__ Bs) {
    for (int i = threadIdx.x; i < 64 * 64; i += 256) {
        const int k = i >> 6;         // W row 0..63
        const int n = i & 63;         // W col 0..63
        Bs[(k >> 1) * 128 + n * 2 + (k & 1)] = W[k * 64 + n];
    }
}

// ---------------------------------------------------------------------------
// K1: HW[j] = hist_embed[j] @ W[64:128, :]   for j in [0, nrows)
// ---------------------------------------------------------------------------
__global__ void __launch_bounds__(256) table_transform_kernel(
    const float* __restrict__ hist_embed,
    const float* __restrict__ Wu, const float* __restrict__ Wi,
    float* __restrict__ HWu, float* __restrict__ HWi, int nrows)
{
    __shared__ float Bs[32 * 128];
    const float* W = (blockIdx.y == 0 ? Wu : Wi) + 64 * 64;   // bottom half of (128,64)
    float* HW = (blockIdx.y == 0 ? HWu : HWi);
    stage_B_frag(W, Bs);
    __syncthreads();

    const int wave = threadIdx.x >> 5;
    const int lane = threadIdx.x & 31;
    const int rowBase = (blockIdx.x * 8 + wave) * 32;
    if (rowBase >= nrows) return;                 // uniform per wave: EXEC all-ones

    const int m = lane & 15;
    const int half = lane >> 4;
    int ar0 = rowBase + m;      if (ar0 >= nrows) ar0 = nrows - 1;
    int ar1 = rowBase + 16 + m; if (ar1 >= nrows) ar1 = nrows - 1;
    const float* A0 = hist_embed + (size_t)ar0 * DIM;
    const float* A1 = hist_embed + (size_t)ar1 * DIM;

    v8f acc0[4], acc1[4];
    #pragma unroll
    for (int i = 0; i < 4; ++i) { acc0[i] = 0.0f; acc1[i] = 0.0f; }

    #pragma unroll
    for (int kt = 0; kt < 16; ++kt) {
        const int k0 = kt * 4 + half * 2;
        const int p  = kt * 2 + half;             // k-pair index
        const v2f a0 = *(const v2f*)(A0 + k0);
        const v2f a1 = *(const v2f*)(A1 + k0);
        #pragma unroll
        for (int nt = 0; nt < 4; ++nt) {
            const int n = nt * 16 + m;
            const v2f b = *(const v2f*)&Bs[p * 128 + n * 2];
            acc0[nt] = __builtin_amdgcn_wmma_f32_16x16x4_f32(
                false, a0, false, b, (short)0, acc0[nt], false, false);
            acc1[nt] = __builtin_amdgcn_wmma_f32_16x16x4_f32(
                false, a1, false, b, (short)0, acc1[nt], false, false);
        }
    }

    #pragma unroll
    for (int nt = 0; nt < 4; ++nt) {
        #pragma unroll
        for (int r = 0; r < 8; ++r) {
            const int r0 = rowBase + r + half * 8;
            const int r1 = r0 + 16;
            if (r0 < nrows) HW[(size_t)r0 * DIM + nt * 16 + m] = acc0[nt][r];
            if (r1 < nrows) HW[(size_t)r1 * DIM + nt * 16 + m] = acc1[nt][r];
        }
    }
}

// ---------------------------------------------------------------------------
// K2: QW[b] = embed[idx[b]] @ W[0:64, :] + bias   (gathered A rows, 4096 rows)
// ---------------------------------------------------------------------------
__global__ void __launch_bounds__(256) q_transform_kernel(
    const int* __restrict__ user_idx, const int* __restrict__ item_idx,
    const float* __restrict__ user_embed, const float* __restrict__ item_embed,
    const float* __restrict__ Wu, const float* __restrict__ bu,
    const float* __restrict__ Wi, const float* __restrict__ bi,
    float* __restrict__ QWu, float* __restrict__ QWi)
{
    __shared__ float Bs[32 * 128];
    __shared__ float bias[64];
    const int mod = blockIdx.y;
    const float* W   = (mod == 0 ? Wu : Wi);                  // top half
    const float* bv  = (mod == 0 ? bu : bi);
    const int*   idx = (mod == 0 ? user_idx : item_idx);
    const float* emb = (mod == 0 ? user_embed : item_embed);
    float*       Q   = (mod == 0 ? QWu : QWi);

    stage_B_frag(W, Bs);
    if (threadIdx.x < 64) bias[threadIdx.x] = bv[threadIdx.x];
    __syncthreads();

    const int wave = threadIdx.x >> 5;
    const int lane = threadIdx.x & 31;
    const int rowBase = (blockIdx.x * 8 + wave) * 32;         // 4096 % 256 == 0
    const int m = lane & 15;
    const int half = lane >> 4;
    const float* A0 = emb + (size_t)idx[rowBase + m] * DIM;
    const float* A1 = emb + (size_t)idx[rowBase + 16 + m] * DIM;

    v8f acc0[4], acc1[4];
    #pragma unroll
    for (int i = 0; i < 4; ++i) { acc0[i] = 0.0f; acc1[i] = 0.0f; }

    #pragma unroll
    for (int kt = 0; kt < 16; ++kt) {
        const int k0 = kt * 4 + half * 2;
        const int p  = kt * 2 + half;
        const v2f a0 = *(const v2f*)(A0 + k0);
        const v2f a1 = *(const v2f*)(A1 + k0);
        #pragma unroll
        for (int nt = 0; nt < 4; ++nt) {
            const int n = nt * 16 + m;
            const v2f b = *(const v2f*)&Bs[p * 128 + n * 2];
            acc0[nt] = __builtin_amdgcn_wmma_f32_16x16x4_f32(
                false, a0, false, b, (short)0, acc0[nt], false, false);
            acc1[nt] = __builtin_amdgcn_wmma_f32_16x16x4_f32(
                false, a1, false, b, (short)0, acc1[nt], false, false);
        }
    }

    #pragma unroll
    for (int nt = 0; nt < 4; ++nt) {
        #pragma unroll
        for (int r = 0; r < 8; ++r) {
            const int r0 = rowBase + r + half * 8;
            const float bb = bias[nt * 16 + m];
            Q[(size_t)r0 * DIM + nt * 16 + m]        = acc0[nt][r] + bb;
            Q[(size_t)(r0 + 16) * DIM + nt * 16 + m] = acc1[nt][r] + bb;
        }
    }
}

// ---------------------------------------------------------------------------
// K3: one block per (batch element, module): scores -> softmax -> lognormal
// reparameterized sample -> renormalize -> KL atomics -> ctx einsum.
// ---------------------------------------------------------------------------
__global__ void __launch_bounds__(256) attention_kernel(
    const int* __restrict__ user_idx, const int* __restrict__ item_idx,
    const int* __restrict__ user_hist, const float* __restrict__ hist_embed,
    const float* __restrict__ HWu, const float* __restrict__ HWi,
    const float* __restrict__ QWu, const float* __restrict__ QWi,
    const float* __restrict__ vu, const float* __restrict__ vi,
    float* __restrict__ ctx_u, float* __restrict__ ctx_i,
    float* __restrict__ acc4)
{
    __shared__ float qw[64];
    __shared__ float vv[64];
    __shared__ int   hidx[L_HIST];
    __shared__ float sc[L_HIST];
    __shared__ float mk[L_HIST];
    __shared__ float red[256];

    const int b   = blockIdx.x;
    const int mod = blockIdx.y;
    const int t    = threadIdx.x;
    const int wave = t >> 5;
    const int lane = t & 31;

    const float* HW  = (mod == 0 ? HWu : HWi);
    const float* QW  = (mod == 0 ? QWu : QWi);
    const float* vvg = (mod == 0 ? vu  : vi );
    float*       ctx = (mod == 0 ? ctx_u : ctx_i);

    const int u    = user_idx[b];
    const int item = item_idx[b];
    if (t < 64) { qw[t] = QW[(size_t)b * DIM + t]; vv[t] = vvg[t]; }
    __syncthreads();

    // ---- phase 1: scores[l] = v . tanh(QW[b] + HW[hist[l]])  (bias folded in QW)
    const float q0 = qw[2 * lane], q1 = qw[2 * lane + 1];
    const float w0 = vv[2 * lane], w1 = vv[2 * lane + 1];
    for (int l = wave; l < L_HIST; l += 8) {
        const int h = user_hist[(size_t)u * L_HIST + l];
        const float* hw = HW + (size_t)h * DIM;
        float p = w0 * tanhf(q0 + hw[2 * lane]) + w1 * tanhf(q1 + hw[2 * lane + 1]);
        #pragma unroll
        for (int off = 16; off > 0; off >>= 1) p += __shfl_down(p, off);
        if (lane == 0) {
            const bool keep = (h != item) && (h != N_ITEMS);
            sc[l]   = keep ? p * (1.0f / TAU) : -1e9f;
            mk[l]   = keep ? 1.0f : 0.0f;
            hidx[l] = h;
        }
    }
    __syncthreads();

    // ---- phase 2: softmax, lognormal sample, renorm, KL ----
    float s = (t < L_HIST) ? sc[t] : -3.0e38f;
    red[t] = s; __syncthreads();
    #pragma unroll
    for (int off = 128; off > 0; off >>= 1) {
        if (t < off) red[t] = fmaxf(red[t], red[t + off]);
        __syncthreads();
    }
    const float mx = red[0]; __syncthreads();

    const float e = (t < L_HIST) ? expf(s - mx) : 0.0f;
    red[t] = e; __syncthreads();
    #pragma unroll
    for (int off = 128; off > 0; off >>= 1) {
        if (t < off) red[t] += red[t + off];
        __syncthreads();
    }
    const float denom = red[0]; __syncthreads();

    float wv = 0.0f, klp = 0.0f, mp = 0.0f;
    if (t < L_HIST) {
        const float phi = e / denom;
        const float mu  = logf(phi + EPSF);
        const float msk = mk[t];
        // KL(LN(mu,0.1) || LN(0,1)) = ln(10) + (0.01 + mu^2)/2 - 0.5
        klp = (2.30258509f + 0.5f * (0.01f + mu * mu) - 0.5f) * msk;
        mp  = msk;
        // deterministic hash RNG -> Box-Muller normal (reparameterized sample)
        const unsigned key = ((unsigned)(b * L_HIST + t)) * 2654435761u ^
                             (mod ? 0x9E3779B9u : 0x85EBCA6Bu);
        const unsigned h1 = mix32(key);
        const unsigned h2 = mix32(key ^ 0x27d4eb2fu);
        const float u1 = ((h1 >> 8) + 1u) * (1.0f / 16777217.0f);
        const float u2 = (h2 >> 8) * (1.0f / 16777216.0f);
        const float nrm = sqrtf(-2.0f * logf(u1)) * cosf(6.28318530718f * u2);
        wv = expf(mu + 0.1f * nrm) * msk;
    }
    red[t] = wv; __syncthreads();
    #pragma unroll
    for (int off = 128; off > 0; off >>= 1) {
        if (t < off) red[t] += red[t + off];
        __syncthreads();
    }
    const float wsum = red[0]; __syncthreads();
    if (t < L_HIST) sc[t] = wv / (wsum + EPSF);

    red[t] = klp; __syncthreads();
    #pragma unroll
    for (int off = 128; off > 0; off >>= 1) {
        if (t < off) red[t] += red[t + off];
        __syncthreads();
    }
    const float klsum = red[0]; __syncthreads();
    red[t] = mp; __syncthreads();
    #pragma unroll
    for (int off = 128; off > 0; off >>= 1) {
        if (t < off) red[t] += red[t + off];
        __syncthreads();
    }
    if (t == 0) {
        atomicAdd(&acc4[2 * mod],     klsum);
        atomicAdd(&acc4[2 * mod + 1], red[0]);
    }
    __syncthreads();

    // ---- phase 3: ctx[d] = sum_l w[l] * hist_embed[hist[l]][d] ----
    const int d = t & 63;
    const int g = t >> 6;
    float a = 0.0f;
    for (int l = g; l < L_HIST; l += 4)
        a += sc[l] * hist_embed[(size_t)hidx[l] * DIM + d];
    red[t] = a; __syncthreads();
    if (g == 0)
        ctx[(size_t)b * DIM + d] = red[d] + red[d + 64] + red[d + 128] + red[d + 192];
}

// ---------------------------------------------------------------------------
// K4: LayerNorm(Qu*ctx_u), LayerNorm(Qi*ctx_i), GMF fuse, logit dot, KL scalar.
// One wave (32 lanes, 2 channels/lane) per batch element.
// ---------------------------------------------------------------------------
__global__ void __launch_bounds__(256) finalize_kernel(
    const int* __restrict__ user_idx, const int* __restrict__ item_idx,
    const float* __restrict__ user_embed, const float* __restrict__ item_embed,
    const float* __restrict__ ctx_u, const float* __restrict__ ctx_i,
    const float* __restrict__ gamma_u, const float* __restrict__ beta_u,
    const float* __restrict__ gamma_i, const float* __restrict__ beta_i,
    const float* __restrict__ pred_w, const float* __restrict__ pred_b,
    const float* __restrict__ acc4, float* __restrict__ out)
{
    const int wave = threadIdx.x >> 5;
    const int lane = threadIdx.x & 31;
    const int b = blockIdx.x * 8 + wave;
    const int d0 = 2 * lane, d1 = 2 * lane + 1;

    const float* qu = user_embed + (size_t)user_idx[b] * DIM;
    const float* qi = item_embed + (size_t)item_idx[b] * DIM;

    // user branch
    float xu0 = qu[d0] * ctx_u[(size_t)b * DIM + d0];
    float xu1 = qu[d1] * ctx_u[(size_t)b * DIM + d1];
    float su = xu0 + xu1;
    #pragma unroll
    for (int off = 16; off > 0; off >>= 1) su += __shfl_xor(su, off);
    const float mu_u = su * (1.0f / 64.0f);
    float du0 = xu0 - mu_u, du1 = xu1 - mu_u;
    float vu_ = du0 * du0 + du1 * du1;
    #pragma unroll
    for (int off = 16; off > 0; off >>= 1) vu_ += __shfl_xor(vu_, off);
    const float rs_u = rsqrtf(vu_ * (1.0f / 64.0f) + 1e-5f);
    const float ur0 = du0 * rs_u * gamma_u[d0] + beta_u[d0];
    const float ur1 = du1 * rs_u * gamma_u[d1] + beta_u[d1];

    // item branch
    float xi0 = qi[d0] * ctx_i[(size_t)b * DIM + d0];
    float xi1 = qi[d1] * ctx_i[(size_t)b * DIM + d1];
    float si = xi0 + xi1;
    #pragma unroll
    for (int off = 16; off > 0; off >>= 1) si += __shfl_xor(si, off);
    const float mu_i = si * (1.0f / 64.0f);
    float di0 = xi0 - mu_i, di1 = xi1 - mu_i;
    float vi_ = di0 * di0 + di1 * di1;
    #pragma unroll
    for (int off = 16; off > 0; off >>= 1) vi_ += __shfl_xor(vi_, off);
    const float rs_i = rsqrtf(vi_ * (1.0f / 64.0f) + 1e-5f);
    const float ir0 = di0 * rs_i * gamma_i[d0] + beta_i[d0];
    const float ir1 = di1 * rs_i * gamma_i[d1] + beta_i[d1];

    float p = ur0 * ir0 * pred_w[d0] + ur1 * ir1 * pred_w[d1];
    #pragma unroll
    for (int off = 16; off > 0; off >>= 1) p += __shfl_xor(p, off);
    if (lane == 0) out[b] = p + pred_b[0];

    if (blockIdx.x == 0 && threadIdx.x == 0) {
        const float klu = BETA * acc4[0] / (acc4[1] + EPSF);
        const float kli = BETA * acc4[2] / (acc4[3] + EPSF);
        out[B_BATCH] = 0.5f * (klu + kli);
    }
}

// ---------------------------------------------------------------------------
extern "C" void kernel_launch(void* const* d_in, const int* in_sizes, int n_in,
                              void* d_out, int out_size, void* d_ws, size_t ws_size,
                              hipStream_t stream)
{
    (void)in_sizes; (void)n_in; (void)out_size; (void)ws_size;
    const int*   user_idx   = (const int*)  d_in[0];
    const int*   item_idx   = (const int*)  d_in[1];
    const int*   user_hist  = (const int*)  d_in[2];
    const float* user_embed = (const float*)d_in[3];
    const float* item_embed = (const float*)d_in[4];
    const float* hist_embed = (const float*)d_in[5];
    const float* Wu = (const float*)d_in[6];
    const float* bu = (const float*)d_in[7];
    const float* vu = (const float*)d_in[8];
    const float* Wi = (const float*)d_in[9];
    const float* bi = (const float*)d_in[10];
    const float* vi = (const float*)d_in[11];
    const float* gamma_u = (const float*)d_in[12];
    const float* beta_u  = (const float*)d_in[13];
    const float* gamma_i = (const float*)d_in[14];
    const float* beta_i  = (const float*)d_in[15];
    const float* pred_w  = (const float*)d_in[16];
    const float* pred_b  = (const float*)d_in[17];
    float* out = (float*)d_out;

    const int    nrows = N_ITEMS + 1;
    const size_t R = (size_t)nrows * DIM;
    const size_t Q = (size_t)B_BATCH * DIM;
    float* ws  = (float*)d_ws;
    float* HWu = ws;
    float* HWi = HWu + R;
    float* QWu = HWi + R;
    float* QWi = QWu + Q;
    float* CXu = QWi + Q;
    float* CXi = CXu + Q;
    float* acc = CXi + Q;

    init_acc_kernel<<<1, 32, 0, stream>>>(acc);

    const int tblocks = (nrows + 255) / 256;   // 32 rows/wave * 8 waves
    table_transform_kernel<<<dim3(tblocks, 2), 256, 0, stream>>>(
        hist_embed, Wu, Wi, HWu, HWi, nrows);

    q_transform_kernel<<<dim3(B_BATCH / 256, 2), 256, 0, stream>>>(
        user_idx, item_idx, user_embed, item_embed, Wu, bu, Wi, bi, QWu, QWi);

    attention_kernel<<<dim3(B_BATCH, 2), 256, 0, stream>>>(
        user_idx, item_idx, user_hist, hist_embed,
        HWu, HWi, QWu, QWi, vu, vi, CXu, CXi, acc);

    finalize_kernel<<<B_BATCH / 8, 256, 0, stream>>>(
        user_idx, item_idx, user_embed, item_embed, CXu, CXi,
        gamma_u, beta_u, gamma_i, beta_i, pred_w, pred_b, acc, out);
}